// AttentionModel_44135083934120
// MI455X (gfx1250) — compile-verified
//
#include <hip/hip_runtime.h>
#include <hip/hip_bf16.h>

#define B_  128
#define N_  1024
#define E_  512
#define H_  8
#define DK_ 64

#define NEG_INF (-__builtin_inff())

typedef __attribute__((ext_vector_type(2))) float v2f;
typedef __attribute__((ext_vector_type(8))) float v8f;

__device__ __forceinline__ float wave_sum32(float v) {
#pragma unroll
  for (int off = 16; off >= 1; off >>= 1) v += __shfl_xor(v, off, 32);
  return v;
}
__device__ __forceinline__ float wave_max32(float v) {
#pragma unroll
  for (int off = 16; off >= 1; off >>= 1) v = fmaxf(v, __shfl_xor(v, off, 32));
  return v;
}

// ---------------------------------------------------------------------------
// Kernel A: per (b,h) attention head. Streams K and V exactly once (coalesced),
// softmax in LDS, writes heads[b, h*DK..] to workspace.
// ---------------------------------------------------------------------------
__global__ __launch_bounds__(256, 2)
void attn_head_kernel(const float* __restrict__ query,
                      const float* __restrict__ K,
                      const float* __restrict__ V,
                      const unsigned char* __restrict__ mask,
                      float* __restrict__ heads /* [B,E] */) {
  const int bh   = blockIdx.x;   // 0..B*H-1
  const int b    = bh >> 3;
  const int h    = bh & 7;
  const int tid  = threadIdx.x;
  const int wave = tid >> 5;
  const int lane = tid & 31;
  const int half = lane >> 4;    // which row of the pair this half-wave handles
  const int l16  = lane & 15;

  __shared__ float q[DK_];
  __shared__ float comp[N_];
  __shared__ float red[8];
  __shared__ float hpart[4][DK_];
  __shared__ float s_inv;

  const float* Kb = K + (size_t)(h * B_ + b) * N_ * DK_;
  const float* Vb = V + (size_t)(h * B_ + b) * N_ * DK_;
  const unsigned char* mb = mask + (size_t)b * N_;

  if (tid < DK_) q[tid] = query[(size_t)b * E_ + h * DK_ + tid];
  __syncthreads();

  // ---- compat[n] = (q . K[n]) / 8, masked.  Two rows per wave-iteration:
  // lanes 0-15 handle row 2p, lanes 16-31 handle row 2p+1; each lane does a
  // float4 chunk, so a wave issues one fully-coalesced 512B load per iter.
  const float4 q4 = ((const float4*)q)[l16];
  for (int p = wave; p < N_ / 2; p += 8) {
    const int n = 2 * p + half;
    __builtin_prefetch(Kb + (size_t)(n + 16) * DK_, 0, 1);
    const float4 k4 = ((const float4*)(Kb + (size_t)n * DK_))[l16];
    float s = fmaf(q4.x, k4.x, fmaf(q4.y, k4.y, fmaf(q4.z, k4.z, q4.w * k4.w)));
#pragma unroll
    for (int off = 8; off >= 1; off >>= 1) s += __shfl_xor(s, off, 16);
    if (l16 == 0) comp[n] = mb[n] ? NEG_INF : s * 0.125f;  // 1/sqrt(64)
  }
  __syncthreads();

  // ---- softmax over comp[0..N) ----
  float lmax = NEG_INF;
  for (int n = tid; n < N_; n += 256) lmax = fmaxf(lmax, comp[n]);
  lmax = wave_max32(lmax);
  if (lane == 0) red[wave] = lmax;
  __syncthreads();
  float gmax = red[0];
#pragma unroll
  for (int i = 1; i < 8; ++i) gmax = fmaxf(gmax, red[i]);
  __syncthreads();

  float lsum = 0.f;
  for (int n = tid; n < N_; n += 256) {
    float e = __expf(comp[n] - gmax);   // masked (-inf) -> 0
    comp[n] = e;
    lsum += e;
  }
  lsum = wave_sum32(lsum);
  if (lane == 0) red[wave] = lsum;
  __syncthreads();
  if (tid == 0) {
    float t = red[0];
#pragma unroll
    for (int i = 1; i < 8; ++i) t += red[i];
    s_inv = 1.0f / t;
  }
  __syncthreads();

  // ---- heads[d] = inv * sum_n p[n] * V[n,d].  Lane-adjacent d => coalesced;
  // groups partition n so every V element is read exactly once.
  const int d = tid & 63;
  const int g = tid >> 6;               // 0..3, owns n-range [g*256, g*256+256)
  const float* Vg = Vb + (size_t)(g * 256) * DK_ + d;
  float acc = 0.f;
  for (int i = 0; i < 256; ++i) {
    acc = fmaf(comp[g * 256 + i], Vg[(size_t)i * DK_], acc);
  }
  hpart[g][d] = acc;
  __syncthreads();
  if (tid < DK_) {
    float s = (hpart[0][tid] + hpart[1][tid] + hpart[2][tid] + hpart[3][tid]) * s_inv;
    heads[(size_t)b * E_ + h * DK_ + tid] = s;
  }
}

// ---------------------------------------------------------------------------
// Kernel B: glimpse[128,512] = heads[128,512] x W_out^T[512,512], exact fp32
// via V_WMMA_F32_16X16X4_F32.  One 16x16 output tile per wave, K=512 in 128
// steps of 4.  A-fragment layout (ISA 7.12.2, 32-bit A 16x4): lanes 0-15 hold
// M=lane with K={0,1}, lanes 16-31 hold M=lane-16 with K={2,3} -> one float2
// per lane per step; B assumed symmetric (N=lane&15).
// ---------------------------------------------------------------------------
__global__ __launch_bounds__(128, 4)
void proj_wmma_kernel(const float* __restrict__ heads,   // [B,E]
                      const float* __restrict__ Wout,    // [E,E] row-major [e][k]
                      float* __restrict__ glimpse) {     // [B,E]
  const int tid  = threadIdx.x;
  const int wave = tid >> 5;
  const int lane = tid & 31;
  const int tile = blockIdx.x * 4 + wave;  // 0..255
  const int b0   = (tile >> 5) * 16;       // 8 M-tiles
  const int e0   = (tile & 31) * 16;       // 32 N-tiles
  const int half = lane >> 4;
  const int l16  = lane & 15;

  const float* Arow = heads + (size_t)(b0 + l16) * E_;  // A: row = batch
  const float* Brow = Wout  + (size_t)(e0 + l16) * E_;  // B^T[k][e] = Wout[e][k]

  v8f c = {0.f, 0.f, 0.f, 0.f, 0.f, 0.f, 0.f, 0.f};
#pragma unroll 4
  for (int k = 0; k < E_; k += 4) {
    const int kk = k + 2 * half;
    const v2f a  = *reinterpret_cast<const v2f*>(Arow + kk);
    const v2f bb = *reinterpret_cast<const v2f*>(Brow + kk);
    c = __builtin_amdgcn_wmma_f32_16x16x4_f32(
        /*neg_a=*/false, a, /*neg_b=*/false, bb,
        /*c_mod=*/(short)0, c, /*reuse_a=*/false, /*reuse_b=*/false);
  }

  // D layout: VGPR i -> (M = i + 8*half, N = l16)
  const int mbase = b0 + half * 8;
#pragma unroll
  for (int i = 0; i < 8; ++i) {
    glimpse[(size_t)(mbase + i) * E_ + e0 + l16] = c[i];
  }
}

// ---------------------------------------------------------------------------
// Kernel C: logits[b,n] = mask ? -inf : 10*tanh((glimpse[b].logit_K[b,n])/sqrt(E))
// One block per b; wave-per-row with coalesced float4 streaming of logit_K.
// ---------------------------------------------------------------------------
__global__ __launch_bounds__(256, 2)
void logits_kernel(const float* __restrict__ glimpse,  // [B,E] (in d_out)
                   const float* __restrict__ logitK,   // [B,N,E]
                   const unsigned char* __restrict__ mask,
                   float* __restrict__ logits) {       // [B,N]
  const int b    = blockIdx.x;
  const int tid  = threadIdx.x;
  const int wave = tid >> 5;
  const int lane = tid & 31;

  __shared__ float g[E_];
  for (int i = tid; i < E_; i += 256) g[i] = glimpse[(size_t)b * E_ + i];
  __syncthreads();

  const float4* g4 = (const float4*)g;
  const float scale = 0.04419417382415922f;  // 1/sqrt(512)
  const unsigned char* mb = mask + (size_t)b * N_;

  for (int n = wave; n < N_; n += 8) {
    const float4* row = (const float4*)(logitK + ((size_t)b * N_ + n) * E_);
    __builtin_prefetch(logitK + ((size_t)b * N_ + n + 8) * E_, 0, 1);
    float acc = 0.f;
#pragma unroll
    for (int j = 0; j < 4; ++j) {
      const float4 x = row[lane + 32 * j];
      const float4 y = g4[lane + 32 * j];
      acc = fmaf(x.x, y.x, fmaf(x.y, y.y, fmaf(x.z, y.z, fmaf(x.w, y.w, acc))));
    }
    acc = wave_sum32(acc);
    if (lane == 0) {
      logits[(size_t)b * N_ + n] = mb[n] ? NEG_INF : tanhf(acc * scale) * 10.0f;
    }
  }
}

// ---------------------------------------------------------------------------
extern "C" void kernel_launch(void* const* d_in, const int* in_sizes, int n_in,
                              void* d_out, int out_size, void* d_ws, size_t ws_size,
                              hipStream_t stream) {
  const float* query          = (const float*)d_in[0];   // [B,1,E]
  const float* glimpse_K      = (const float*)d_in[1];   // [H,B,1,N,DK]
  const float* glimpse_V      = (const float*)d_in[2];   // [H,B,1,N,DK]
  const float* logit_K        = (const float*)d_in[3];   // [B,1,N,E]
  const float* W_out          = (const float*)d_in[4];   // [E,E]
  const unsigned char* mask   = (const unsigned char*)d_in[5];  // [B,1,N] bool

  float* logits  = (float*)d_out;             // [B,N]
  float* glimpse = (float*)d_out + B_ * N_;   // [B,E]
  float* heads   = (float*)d_ws;              // [B,E] scratch

  attn_head_kernel<<<B_ * H_, 256, 0, stream>>>(query, glimpse_K, glimpse_V, mask, heads);
  proj_wmma_kernel<<<64, 128, 0, stream>>>(heads, W_out, glimpse);
  logits_kernel<<<B_, 256, 0, stream>>>(glimpse, logit_K, mask, logits);
}